// DeepSurvLoss_41867341201488
// MI455X (gfx1250) — compile-verified
//
#include <hip/hip_runtime.h>
#include <hip/hip_bf16.h>
#include <math.h>

// CDNA5 / gfx1250: wave32, WMMA f32 16x16x4 (full-precision matrix pipe).
typedef __attribute__((ext_vector_type(2))) float v2f;
typedef __attribute__((ext_vector_type(8))) float v8f;

#define EPS_F 1e-6f
#define TILE_J 1024

// ---------------------------------------------------------------------------
// Kernel 1: elementwise precompute of the value vectors.
//   expR[j]  = exp(P_risk[j])
//   expSE[j] = exp(P_surv[j]) * E[j]
//   Ef[j]    = (float)E[j]
// ---------------------------------------------------------------------------
__global__ void cox_precompute(const float* __restrict__ Pr,
                               const float* __restrict__ Ps,
                               const int*   __restrict__ E,
                               float* __restrict__ expR,
                               float* __restrict__ expSE,
                               float* __restrict__ Ef,
                               int n) {
    int i = blockIdx.x * blockDim.x + threadIdx.x;
    if (i < n) {
        float e = (float)E[i];
        expR[i]  = expf(Pr[i]);
        expSE[i] = expf(Ps[i]) * e;
        Ef[i]    = e;
    }
}

// ---------------------------------------------------------------------------
// Kernel 2: masked risk-set sums via V_WMMA_F32_16X16X4_F32.
// One wave (32 lanes) per block; block b owns rows i0 = 16*b .. 16*b+15.
//
// A (16x4 f32, 2 VGPRs): mask tile. ISA layout: lanes 0-15 hold (M=lane, K=0/1)
// in v0/v1; lanes 16-31 hold (M=lane-16, K=2/3).
// B (4x16 f32, 2 VGPRs): column 0 = value vector, column 1 = count vector,
// columns 2-15 = 0. K split across lane halves mirrors A (K={0,1} low half,
// K={2,3} high half; v0=first K of half, v1=second).
// C/D (16x16 f32, 8 VGPRs): v[r], lanes 0-15 -> (M=r, N=lane);
// lanes 16-31 -> (M=r+8, N=lane-16).
// ---------------------------------------------------------------------------
__global__ __launch_bounds__(32)
void cox_wmma(const float* __restrict__ T,
              const float* __restrict__ Pr,
              const float* __restrict__ Ps,
              const int*   __restrict__ E,
              const float* __restrict__ expR,
              const float* __restrict__ expSE,
              const float* __restrict__ Ef,
              float* __restrict__ partials,
              int n) {
    __shared__ float sT[TILE_J];
    __shared__ float sER[TILE_J];
    __shared__ float sES[TILE_J];
    __shared__ float sEF[TILE_J];
    __shared__ float sRed[16 * 4];

    const int lane = threadIdx.x;          // 0..31
    const int i0   = blockIdx.x * 16;
    const int nidx = lane & 15;            // N column / M row within half
    const int kb   = (lane >> 4) << 1;     // 0 for lanes 0-15, 2 for 16-31

    const float Ti = T[i0 + nidx];         // row key for this lane's A rows
    const float c0 = (nidx == 0) ? 1.0f : 0.0f;  // this lane is B column 0
    const float c1 = (nidx == 1) ? 1.0f : 0.0f;  // this lane is B column 1

    v8f accR = {};   // risk:  col0 = sum mask_less * expR, col1 = count_less
    v8f accS = {};   // surv:  col0 = sum mask_grt * expSE, col1 = sum mask_grt*Ef

    for (int jt = 0; jt < n; jt += TILE_J) {
        // Cooperative stage of the j-tile into LDS (uniform trip count: no
        // divergence, EXEC stays all-ones for the WMMAs below).
        for (int t = lane; t < TILE_J; t += 32) {
            sT[t]  = T[jt + t];
            sER[t] = expR[jt + t];
            sES[t] = expSE[jt + t];
            sEF[t] = Ef[jt + t];
        }
        __syncthreads();

        for (int jc = 0; jc < TILE_J; jc += 4) {
            const int j0 = jc + kb;        // 8B aligned (jc%4==0, kb even)
            const v2f tj = *reinterpret_cast<const v2f*>(&sT[j0]);
            const v2f er = *reinterpret_cast<const v2f*>(&sER[j0]);
            const v2f es = *reinterpret_cast<const v2f*>(&sES[j0]);
            const v2f ef = *reinterpret_cast<const v2f*>(&sEF[j0]);

            v2f aL, aG, bR, bS;
            // Strict comparisons => exact tie semantics of the reference.
            aL.x = (Ti < tj.x) ? 1.0f : 0.0f;
            aL.y = (Ti < tj.y) ? 1.0f : 0.0f;
            aG.x = (Ti > tj.x) ? 1.0f : 0.0f;
            aG.y = (Ti > tj.y) ? 1.0f : 0.0f;
            // B columns, branchless: col0 = value, col1 = count weight.
            bR.x = fmaf(c0, er.x, c1);
            bR.y = fmaf(c0, er.y, c1);
            bS.x = fmaf(c0, es.x, c1 * ef.x);
            bS.y = fmaf(c0, es.y, c1 * ef.y);

            accR = __builtin_amdgcn_wmma_f32_16x16x4_f32(
                false, aL, false, bR, (short)0, accR, false, false);
            accS = __builtin_amdgcn_wmma_f32_16x16x4_f32(
                false, aG, false, bS, (short)0, accS, false, false);
        }
        __syncthreads();
    }

    // Extract D columns 0 and 1 for all 16 rows through LDS.
    const int mhalf = (lane >> 4) * 8;
    if (nidx < 2) {
#pragma unroll
        for (int r = 0; r < 8; ++r) {
            sRed[(mhalf + r) * 4 + nidx]     = accR[r];  // risk sum / count
            sRed[(mhalf + r) * 4 + 2 + nidx] = accS[r];  // surv sum / count
        }
    }
    __syncthreads();

    float cR = 0.0f, wR = 0.0f, cS = 0.0f, wS = 0.0f;
    if (lane < 16) {
        const int i = i0 + lane;
        const float sumR = sRed[lane * 4 + 0];
        const float cntR = sRed[lane * 4 + 1];
        const float sumS = sRed[lane * 4 + 2];
        const float cntS = sRed[lane * 4 + 3];
        wR = (cntR > 0.0f) ? (float)E[i] : 0.0f;   // E_risk
        wS = (cntS > 0.0f) ? 1.0f : 0.0f;          // E_surv
        // log(exp(P)/(S+eps)) = P - log(S+eps)
        cR = wR * (Pr[i] - logf(sumR + EPS_F));
        cS = wS * (Ps[i] - logf(sumS + EPS_F));
    }
    // Wave reduction (lanes >=16 contribute zeros).
#pragma unroll
    for (int off = 16; off >= 1; off >>= 1) {
        cR += __shfl_down(cR, off, 32);
        wR += __shfl_down(wR, off, 32);
        cS += __shfl_down(cS, off, 32);
        wS += __shfl_down(wS, off, 32);
    }
    if (lane == 0) {
        float* p = partials + blockIdx.x * 4;
        p[0] = cR;  p[1] = wR;  p[2] = cS;  p[3] = wS;
    }
}

// ---------------------------------------------------------------------------
// Kernel 3: deterministic fixed-order reduction of per-block partials.
// ---------------------------------------------------------------------------
__global__ void cox_finalize(const float* __restrict__ partials, int nblk,
                             float* __restrict__ out) {
    __shared__ float s[4];
    const int lane = threadIdx.x;
    if (lane < 4) {
        float acc = 0.0f;
        for (int t = 0; t < nblk; ++t) acc += partials[t * 4 + lane];
        s[lane] = acc;
    }
    __syncthreads();
    if (lane == 0) {
        out[0] = -(s[0] / s[1]);   // loss_risk
        out[1] = -(s[2] / s[3]);   // loss_surv
    }
}

// ---------------------------------------------------------------------------
extern "C" void kernel_launch(void* const* d_in, const int* in_sizes, int n_in,
                              void* d_out, int out_size, void* d_ws, size_t ws_size,
                              hipStream_t stream) {
    (void)n_in; (void)out_size; (void)ws_size;
    const float* Pr = (const float*)d_in[0];
    const float* Ps = (const float*)d_in[1];
    const float* T  = (const float*)d_in[2];
    const int*   E  = (const int*)d_in[3];
    const int n = in_sizes[0];              // 8192
    float* out = (float*)d_out;

    float* expR     = (float*)d_ws;         // [n]
    float* expSE    = expR + n;             // [n]
    float* Ef       = expSE + n;            // [n]
    float* partials = Ef + n;               // [4 * (n/16)]
    const int nblk = n / 16;

    cox_precompute<<<(n + 255) / 256, 256, 0, stream>>>(Pr, Ps, E,
                                                        expR, expSE, Ef, n);
    cox_wmma<<<nblk, 32, 0, stream>>>(T, Pr, Ps, E, expR, expSE, Ef,
                                      partials, n);
    cox_finalize<<<1, 32, 0, stream>>>(partials, nblk, out);
}